// ResonantFFN_75892072121095
// MI455X (gfx1250) — compile-verified
//
#include <hip/hip_runtime.h>
#include <hip/hip_bf16.h>
#include <math.h>

#define D_MODEL 1024
#define NNEUR   2048
#define BATCH   64
#define LUT     4096
#define LUT_MASK (LUT - 1)
// LUT / (2*pi)
#define POS_SCALE 651.8986469044033f

typedef float v2f __attribute__((ext_vector_type(2)));
typedef float v8f __attribute__((ext_vector_type(8)));

// ---------------------------------------------------------------------------
// Kernel 0: transpose x (B, D) -> xT (D, B) so phase-A lanes (mapped to batch)
// get coalesced 128B loads per d step.
// ---------------------------------------------------------------------------
__global__ __launch_bounds__(256) void transpose_x_kernel(
    const float* __restrict__ x, float* __restrict__ xT) {
  int i = blockIdx.x * 256 + threadIdx.x;      // 65536 elements
  int b = i >> 10;                             // / D_MODEL
  int d = i & (D_MODEL - 1);
  xT[d * BATCH + b] = x[i];
}

// ---------------------------------------------------------------------------
// Kernel 1 (dominant): per (b, n) accumulate sum_d sin(theta), cos(theta).
// Block = 256 threads (8 waves), 4 neurons per block (2 waves per neuron:
// wave parity selects batch half). Each lane owns one (b, n) accumulator pair
// -> no cross-lane reduction. sin/cos LUT lives in LDS as float2 so each
// table index is one ds_load_b64 gather. W / B_p rows stream as float4
// (global_load_b128) -> 6 vmem + 8 ds per 4 thetas.
// ---------------------------------------------------------------------------
__global__ __launch_bounds__(256) void resonant_sum_kernel(
    const float* __restrict__ W, const float* __restrict__ Bp,
    const float* __restrict__ xT, const float* __restrict__ t,
    const float* __restrict__ sinT, const float* __restrict__ cosT,
    float* __restrict__ Csum, float* __restrict__ Ssum) {
  __shared__ float2 lut[LUT];                  // 32 KB of the 320 KB WGP LDS
  for (int i = threadIdx.x; i < LUT; i += 256) {
    lut[i] = make_float2(sinT[i], cosT[i]);
  }
  __syncthreads();

  const int wave = threadIdx.x >> 5;
  const int lane = threadIdx.x & 31;
  const int n = blockIdx.x * 4 + (wave >> 1);      // 4 neurons / block
  const int b = lane + 32 * (wave & 1);            // which batch half

  const float tb = t[b];
  const float4* __restrict__ wrow =
      (const float4*)(W + (size_t)n * D_MODEL);    // 16B-aligned broadcast rows
  const float4* __restrict__ brow =
      (const float4*)(Bp + (size_t)n * D_MODEL);
  const float* __restrict__ xcol = xT + b;         // coalesced per-lane stream

  float c_acc0 = 0.0f, s_acc0 = 0.0f;
  float c_acc1 = 0.0f, s_acc1 = 0.0f;              // dual accumulators for ILP

  for (int d4 = 0; d4 < D_MODEL / 4; ++d4) {
    float4 w4  = wrow[d4];                         // global_load_b128
    float4 bp4 = brow[d4];                         // global_load_b128
    float wv[4]  = {w4.x, w4.y, w4.z, w4.w};
    float bpv[4] = {bp4.x, bp4.y, bp4.z, bp4.w};
#pragma unroll
    for (int j = 0; j < 4; ++j) {
      float xv = xcol[(d4 * 4 + j) * BATCH];       // global_load_b32, coalesced
      // 1/(1 + |w|) via v_rcp_f32 (tolerance of LUT interp >> rcp error)
      float invw  = __builtin_amdgcn_rcpf(1.0f + __builtin_fabsf(wv[j]));
      float theta = __builtin_fmaf(xv, invw, bpv[j] + tb);
      float pos   = theta * POS_SCALE;
      float fb    = floorf(pos);
      float frac  = pos - fb;                      // [0,1)
      int   i0    = ((int)fb) & LUT_MASK;          // mod LUT (power of two)
      int   i1    = (i0 + 1) & LUT_MASK;
      float2 sc0  = lut[i0];                       // ds_load_b64
      float2 sc1  = lut[i1];                       // ds_load_b64
      float s = __builtin_fmaf(frac, sc1.x - sc0.x, sc0.x);
      float c = __builtin_fmaf(frac, sc1.y - sc0.y, sc0.y);
      if (j & 1) { s_acc1 += s; c_acc1 += c; }
      else       { s_acc0 += s; c_acc0 += c; }
    }
  }

  Csum[(size_t)b * NNEUR + n] = c_acc0 + c_acc1;
  Ssum[(size_t)b * NNEUR + n] = s_acc0 + s_acc1;
}

// ---------------------------------------------------------------------------
// Kernel 2: out = silu(C @ Pr^T + S @ Pi^T) with V_WMMA_F32_16X16X4_F32.
// One wave per 16x16 output tile: bt in [0,4) x mt in [0,64). K streams by 4;
// two WMMAs per K-step (real + imag) chained into one f32 accumulator.
// A (16x4 f32) frag layout: lanes 0-15 -> K=0,1 ; lanes 16-31 -> K=2,3
// => per-lane float2 load at row r, col (k + 2*(lane>>4)). B mirrors it.
// ---------------------------------------------------------------------------
__global__ __launch_bounds__(128) void proj_silu_wmma_kernel(
    const float* __restrict__ Csum, const float* __restrict__ Ssum,
    const float* __restrict__ Pr, const float* __restrict__ Pi,
    float* __restrict__ out) {
  const int wid  = threadIdx.x >> 5;           // 4 waves per block
  const int lane = threadIdx.x & 31;
  const int bt   = wid;                        // b tile 0..3
  const int mt   = blockIdx.x;                 // m tile 0..63
  const int r    = lane & 15;
  const int half = lane >> 4;
  const int koff = half << 1;                  // 0 or 2
  const int b0   = bt * 16;
  const int m0   = mt * 16;

  const float* __restrict__ aC = Csum + (size_t)(b0 + r) * NNEUR + koff;
  const float* __restrict__ aS = Ssum + (size_t)(b0 + r) * NNEUR + koff;
  const float* __restrict__ bR = Pr + (size_t)(m0 + r) * NNEUR + koff;
  const float* __restrict__ bI = Pi + (size_t)(m0 + r) * NNEUR + koff;

  v8f acc = {};
#pragma unroll 4
  for (int k = 0; k < NNEUR; k += 4) {
    v2f ar = *(const v2f*)(aC + k);            // global_load_b64, 8B aligned
    v2f br = *(const v2f*)(bR + k);
    acc = __builtin_amdgcn_wmma_f32_16x16x4_f32(
        false, ar, false, br, (short)0, acc, false, false);
    v2f ai = *(const v2f*)(aS + k);
    v2f bi = *(const v2f*)(bI + k);
    acc = __builtin_amdgcn_wmma_f32_16x16x4_f32(
        false, ai, false, bi, (short)0, acc, false, false);
  }

  // D layout: VGPR v -> rows M=v (lanes 0-15) and M=v+8 (lanes 16-31), N=lane&15
#pragma unroll
  for (int v = 0; v < 8; ++v) {
    float z = acc[v];
    // SiLU with v_rcp_f32: denominator is in [1, 2] for z>=0 and (1, inf) else,
    // strictly positive -> fast reciprocal is safe (avoids IEEE div sequence).
    float s = z * __builtin_amdgcn_rcpf(1.0f + __expf(-z));
    out[(size_t)(b0 + v + 8 * half) * D_MODEL + m0 + r] = s;
  }
}

// ---------------------------------------------------------------------------
extern "C" void kernel_launch(void* const* d_in, const int* in_sizes, int n_in,
                              void* d_out, int out_size, void* d_ws, size_t ws_size,
                              hipStream_t stream) {
  const float* x    = (const float*)d_in[0];   // (64, 1024)
  const float* t    = (const float*)d_in[1];   // (64,)
  const float* W    = (const float*)d_in[2];   // (2048, 1024)
  const float* Bp   = (const float*)d_in[3];   // (2048, 1024)
  const float* Pr   = (const float*)d_in[4];   // (1024, 2048)
  const float* Pi   = (const float*)d_in[5];   // (1024, 2048)
  const float* sinT = (const float*)d_in[6];   // (4096,)
  const float* cosT = (const float*)d_in[7];   // (4096,)
  float* out = (float*)d_out;                  // (64, 1024)

  float* xT   = (float*)d_ws;                  // 65536 f32
  float* Csum = xT + (size_t)D_MODEL * BATCH;  // 131072 f32
  float* Ssum = Csum + (size_t)BATCH * NNEUR;  // 131072 f32

  transpose_x_kernel<<<(BATCH * D_MODEL) / 256, 256, 0, stream>>>(x, xT);
  resonant_sum_kernel<<<NNEUR / 4, 256, 0, stream>>>(W, Bp, xT, t, sinT, cosT,
                                                     Csum, Ssum);
  proj_silu_wmma_kernel<<<D_MODEL / 16, 128, 0, stream>>>(Csum, Ssum, Pr, Pi, out);
}